// Downsampler_47966194762291
// MI455X (gfx1250) — compile-verified
//
#include <hip/hip_runtime.h>
#include <math.h>
#include <stdint.h>

typedef float v2f __attribute__((ext_vector_type(2)));
typedef float v8f __attribute__((ext_vector_type(8)));

#define HB 512
#define WB 512
#define HS 256
#define WS 256
#define K2 9
#define NPB (HS*WS*K2)     /* 589824 samples per batch          */
#define ROWSZ (K2*WS)      /* 2304 samples per (b,i) pixel row  */
#define TPB 128

__global__ __launch_bounds__(TPB) void downsampler_kernel(
    const float* __restrict__ img,
    const float* __restrict__ kern,
    const float* __restrict__ offh,
    const float* __restrict__ offv,
    float* __restrict__ out)
{
  __shared__ float sW[TPB][12];        /*  6 KB: tap weights, K padded to 12 */
  __shared__ float sV[3][TPB][12];     /* 18 KB: sampled values per channel  */
  __shared__ float sD[TPB/32][32][8];  /*  4 KB: per-wave WMMA D staging     */
  __shared__ float nbrH[K2][WS];       /*  9 KB: neighbor offsets_h row      */
  __shared__ float nbrV[K2][WS];       /*  9 KB: neighbor offsets_v row      */

  const int tid  = threadIdx.x;
  const int lane = tid & 31;
  const int wv   = tid >> 5;

  const int pix = blockIdx.x * TPB + tid;  /* global output pixel (b,i,j) */
  const int b   = pix >> 16;
  const int ij  = pix & 0xFFFF;
  const int i   = ij >> 8;
  const int j   = ij & 0xFF;

  /* reference broadcast quirk: base coordinate uses the COLUMN index j */
  const float bx = (float)j + 0.5f;

  const float* ohB = offh + (size_t)b * NPB;
  const float* ovB = offv + (size_t)b * NPB;
  const float* knB = kern + (size_t)b * NPB;
  const float* imB = img  + (size_t)b * 3 * HB * WB;

  /* ---- phase 0: async-stage the neighbor offset row into LDS ----
     neighbor samples n'=2n+t of this block cover exactly one pixel row:
     rowN = ((2i) & 255) + (j>=128), uniform over the block.            */
  const int rowN = ((2 * i) & 255) + ((j >> 7) & 1);
  {
    const float* srcH = ohB + (size_t)rowN * WS;   /* + k*HS*WS per plane */
    const float* srcV = ovB + (size_t)rowN * WS;
    for (int u = tid; u < K2 * (WS / 4); u += TPB) {  /* 16B chunks       */
      const int plane = u >> 6;                        /* 64 chunks/plane */
      const int c16   = u & 63;
      const float* gh = srcH + (size_t)plane * (HS*WS) + c16 * 4;
      const float* gv = srcV + (size_t)plane * (HS*WS) + c16 * 4;
      const uint32_t lh = (uint32_t)(uintptr_t)&nbrH[plane][c16 * 4];
      const uint32_t lv = (uint32_t)(uintptr_t)&nbrV[plane][c16 * 4];
      asm volatile("global_load_async_to_lds_b128 %0, %1, off"
                   :: "v"(lh), "v"(gh) : "memory");
      asm volatile("global_load_async_to_lds_b128 %0, %1, off"
                   :: "v"(lv), "v"(gv) : "memory");
    }
    asm volatile("s_wait_asynccnt 0x0" ::: "memory");
  }
  __syncthreads();

  /* ---- phase 1: per-pixel taps -> LDS ---- */
  #pragma unroll 1
  for (int k = 0; k < K2; ++k) {
    const int planeIdx = k * (HS*WS) + i * WS + j;
    const float oh = ohB[planeIdx];
    const float ov = ovB[planeIdx];
    const float cx = ((oh + 1.5f) + (float)(k / 3)) + bx;
    const float cy = ((ov + 1.5f) + (float)(k % 3)) + bx;
    int x0 = (int)floorf(cx); int x1 = x0 + 1;
    int y0 = (int)floorf(cy); int y1 = y0 + 1;
    x0 = min(max(x0, 0), WB-1); x1 = min(max(x1, 0), WB-1);
    y0 = min(max(y0, 0), HB-1); y1 = min(max(y1, 0), HB-1);
    const float scl = 1.0f / (float)((x1 - x0) * (y1 - y0));

    /* single gathered point: V_c = img[b,c,x0,y0] */
    const int p0 = x0 * WB + y0;
    const float V0 = imB[p0];
    const float V1 = imB[HB*WB + p0];
    const float V2 = imB[2*HB*WB + p0];

    /* scrambled weights: neighbor samples n'=2n+t, served from LDS */
    const int n = (i * WS + j) * K2 + k;
    float f0, f1, g0, g1;
    #pragma unroll
    for (int t = 0; t < 2; ++t) {
      const int col  = 2*n + t;
      const bool useA = (col < NPB);
      const int np = useA ? col : (col - NPB);
      const int rp = np - rowN * ROWSZ;      /* offset within staged row */
      const int jp = rp / K2;
      const int kp = rp - jp * K2;
      const float ohp = nbrH[kp][jp];
      const float ovp = nbrV[kp][jp];
      const float bp  = (float)jp + 0.5f;
      const float cxp = ((ohp + 1.5f) + (float)(kp / 3)) + bp;
      const float cyp = ((ovp + 1.5f) + (float)(kp % 3)) + bp;
      int xa = (int)floorf(cxp); int xb = xa + 1;
      int ya = (int)floorf(cyp); int yb = ya + 1;
      xa = min(max(xa, 0), WB-1); xb = min(max(xb, 0), WB-1);
      ya = min(max(ya, 0), HB-1); yb = min(max(yb, 0), HB-1);
      const float fv = useA ? ((float)xb - cxp) : (cxp - (float)xa);
      const float gv = useA ? ((float)yb - cyp) : (cyp - (float)ya);
      if (t == 0) { f0 = fv; g0 = gv; } else { f1 = fv; g1 = gv; }
    }

    /* res[r] = scl * sum_{p,q} f[q] g[p] V[cidx(p,q,r)] (collapsed) */
    const float r0 = scl * (g0*(f0+f1)*V0      + g1*(f0*V1 + f1*V2));
    const float r1 = scl * (g0*(f0*V0 + f1*V1) + g1*(f0*V1 + f1*V2));
    const float r2 = scl * (g0*(f0*V0 + f1*V1) + g1*(f0+f1)*V2);

    sW[tid][k]    = knB[planeIdx];
    sV[0][tid][k] = r0;
    sV[1][tid][k] = r1;
    sV[2][tid][k] = r2;
  }
  #pragma unroll
  for (int k = K2; k < 12; ++k) {        /* zero K padding */
    sW[tid][k] = 0.0f;
    sV[0][tid][k] = 0.0f; sV[1][tid][k] = 0.0f; sV[2][tid][k] = 0.0f;
  }
  __syncthreads();

  /* ---- phase 2: 9-tap reduction via V_WMMA_F32_16X16X4_F32 ----
     D = W(16xK) * V^T(Kx16); diag(D)[p] = sum_k w[p,k] v[p,k,c].      */
  float o3[3] = {0.f, 0.f, 0.f};
  const int half = lane >> 4;      /* K-pair selector within a K=4 chunk */
  const int pl   = lane & 15;      /* M / N index                        */
  #pragma unroll 1
  for (int g2 = 0; g2 < 2; ++g2) {
    const int pb = wv * 32 + g2 * 16;   /* first pixel of this 16-group */
    #pragma unroll 1
    for (int c = 0; c < 3; ++c) {
      v8f acc = {0.f,0.f,0.f,0.f,0.f,0.f,0.f,0.f};
      #pragma unroll
      for (int kc = 0; kc < 3; ++kc) {
        const int kidx = 4*kc + (half << 1);
        v2f a, bb;
        a[0]  = sW[pb + pl][kidx];
        a[1]  = sW[pb + pl][kidx + 1];
        bb[0] = sV[c][pb + pl][kidx];
        bb[1] = sV[c][pb + pl][kidx + 1];
        acc = __builtin_amdgcn_wmma_f32_16x16x4_f32(false, a, false, bb,
                                                    (short)0, acc, false, false);
      }
      #pragma unroll
      for (int r = 0; r < 8; ++r) sD[wv][lane][r] = acc[r];
      __syncthreads();
      /* diagonal: p<8 -> (lane p, vgpr p); p>=8 -> (lane p+16, vgpr p-8) */
      const int srcLane = (pl < 8) ? pl : (pl + 16);
      const float dv = sD[wv][srcLane][pl & 7];
      if (half == g2) o3[c] = dv;
      __syncthreads();
    }
  }

  /* ---- epilogue: *255 and sine correction, layout (B,h,w,C) ---- */
  const float TWO_PI = 6.28318530717958647692f;
  #pragma unroll
  for (int c = 0; c < 3; ++c) {
    const float o = o3[c] * 255.0f;
    out[(size_t)pix * 3 + c] = o - sinf(TWO_PI * o) / TWO_PI;
  }
}

extern "C" void kernel_launch(void* const* d_in, const int* in_sizes, int n_in,
                              void* d_out, int out_size, void* d_ws, size_t ws_size,
                              hipStream_t stream) {
  (void)in_sizes; (void)n_in; (void)d_ws; (void)ws_size; (void)out_size;
  const float* img  = (const float*)d_in[0];
  const float* kern = (const float*)d_in[1];
  const float* offh = (const float*)d_in[2];
  const float* offv = (const float*)d_in[3];
  float* out = (float*)d_out;

  const int numPix = 8 * HS * WS;            /* 524288 */
  dim3 grid(numPix / TPB);                   /* 4096 blocks, 128 thr = 4 waves */
  downsampler_kernel<<<grid, TPB, 0, stream>>>(img, kern, offh, offv, out);
}